// ScaleQuadraticDilation_33131377722048
// MI455X (gfx1250) — compile-verified
//
#include <hip/hip_runtime.h>
#include <stdint.h>

// Tensor geometry: x, out are (B, H, W, S, C) f32
#define B_ 8
#define H_ 256
#define W_ 256
#define S_ 3
#define C_ 32
#define STRIDE_W 96        // S_*C_
#define STRIDE_H 24576     // W_*STRIDE_W
#define STRIDE_B 6291456   // H_*STRIDE_H

// gfx1250 async global->LDS path (gated so compile never breaks if absent)
#if defined(__HIP_DEVICE_COMPILE__) && __has_builtin(__builtin_amdgcn_global_load_async_to_lds_b32) && __has_builtin(__builtin_amdgcn_s_wait_asynccnt)
#define ASYNC_LDS 1
#else
#define ASYNC_LDS 0
#endif

typedef __attribute__((address_space(1))) int* gptr_i32;
typedef __attribute__((address_space(3))) int* lptr_i32;

// fmax(fmax(a,b),c) folds to v_max3_num_f32 on gfx1250 (exact: same value set).
__device__ __forceinline__ float max3f(float a, float b, float c) {
  return fmaxf(fmaxf(a, b), c);
}
__device__ __forceinline__ float min3f(float a, float b, float c) {
  return fminf(fminf(a, b), c);
}

// Order-preserving mapping float -> uint32 (ascending), for deterministic atomic min.
__device__ __forceinline__ uint32_t f32_order_key(float f) {
  uint32_t u = __float_as_uint(f);
  return (u & 0x80000000u) ? ~u : (u | 0x80000000u);
}
__device__ __forceinline__ float f32_order_decode(uint32_t u) {
  return (u & 0x80000000u) ? __uint_as_float(u ^ 0x80000000u)
                           : __uint_as_float(~u);
}

__global__ void init_mins_kernel(uint32_t* mins) {
  if (threadIdx.x < S_) mins[threadIdx.x] = 0xFFFFFFFFu;  // encodes +max
}

// Global min of each scale slice x[..., s, :], s = 0..2, via float4 streaming.
__global__ __launch_bounds__(256) void slice_min_kernel(const float4* __restrict__ x4,
                                                        uint32_t* __restrict__ mins) {
  constexpr int NPIX = B_ * H_ * W_;
  constexpr int N4 = NPIX * (C_ / 4);  // float4 elements per scale slice
  float m0 = 3.402823466e38f, m1 = m0, m2 = m0;
  const int stride = gridDim.x * blockDim.x;
  for (int j = blockIdx.x * blockDim.x + threadIdx.x; j < N4; j += stride) {
    const int p = j >> 3;              // pixel index
    const int q = j & 7;               // which float4 within 32 channels
    const size_t base = (size_t)p * (STRIDE_W / 4) + q;
    float4 v0 = x4[base];              // scale 0
    float4 v1 = x4[base + 8];          // scale 1 (+32 floats)
    float4 v2 = x4[base + 16];         // scale 2 (+64 floats)
    m0 = min3f(m0, fminf(v0.x, v0.y), fminf(v0.z, v0.w));
    m1 = min3f(m1, fminf(v1.x, v1.y), fminf(v1.z, v1.w));
    m2 = min3f(m2, fminf(v2.x, v2.y), fminf(v2.z, v2.w));
  }
  __shared__ float red0[256], red1[256], red2[256];
  red0[threadIdx.x] = m0; red1[threadIdx.x] = m1; red2[threadIdx.x] = m2;
  __syncthreads();
  for (int off = 128; off > 0; off >>= 1) {
    if ((int)threadIdx.x < off) {
      red0[threadIdx.x] = fminf(red0[threadIdx.x], red0[threadIdx.x + off]);
      red1[threadIdx.x] = fminf(red1[threadIdx.x], red1[threadIdx.x + off]);
      red2[threadIdx.x] = fminf(red2[threadIdx.x], red2[threadIdx.x + off]);
    }
    __syncthreads();
  }
  if (threadIdx.x == 0) {
    atomicMin((unsigned int*)&mins[0], (unsigned int)f32_order_key(red0[0]));
    atomicMin((unsigned int*)&mins[1], (unsigned int)f32_order_key(red1[0]));
    atomicMin((unsigned int*)&mins[2], (unsigned int)f32_order_key(red2[0]));
  }
}

// Fused separable max-plus dilation for one scale.
// Block = (h = blockIdx.x, b = blockIdx.y). 256 threads = 8 waves.
// Lane = channel (wave32 == 32 channels). Wave 'grp' owns columns w = grp, grp+8, ...
// Exactness notes:
//  * pad columns: max_d(mn+kv[d]) == mn (center tap kv[WR] == -0.0), so T = mn.
//  * out-of-range H taps contribute mn+kv[d] <= mn <= x[h][w]+kv[WR] (center tap,
//    always in range) -> dominated, safe to skip; acc may be seeded with mn.
//  * max is associative/commutative (no NaNs) -> accumulator trees / max3 exact.
template <int WR, int SCALE>
__global__ __launch_bounds__(256) void dilate2d_kernel(const float* __restrict__ x,
                                                       const float* __restrict__ coef,
                                                       const uint32_t* __restrict__ mins,
                                                       float* __restrict__ out) {
  constexpr int L = 2 * WR + 1;                         // odd
  constexpr int WP = W_ + 2 * WR;                       // padded width
  constexpr int SV = 2 * ((1 << (SCALE + 1)) - 1);      // 2, 6, 14
  constexpr float S2 = (float)(SV * SV);                // 4, 36, 196

  __shared__ float T[WP * C_];
#if ASYNC_LDS
  __shared__ float coefLDS[C_];
#endif

  const int tid = threadIdx.x;
  const int c = tid & (C_ - 1);
  const int grp = tid >> 5;
  const int h = blockIdx.x;
  const int b = blockIdx.y;

#if ASYNC_LDS
  if (tid < C_) {
    __builtin_amdgcn_global_load_async_to_lds_b32(
        (gptr_i32)(coef + tid), (lptr_i32)(&coefLDS[tid]), 0, 0);
  }
  __builtin_amdgcn_s_wait_asynccnt(0);
  __syncthreads();
  const float coefc = coefLDS[c];
#else
  const float coefc = coef[c];
#endif

  const float mn = f32_order_decode(mins[SCALE]);

  // kv[d] = (-(d-WR)^2 / s^2) * coef[c]; constant part folded in f32, then one
  // rounded multiply — matches numpy's  k = filt[:,None] * coef  exactly.
  float kv[L];
#pragma unroll
  for (int d = 0; d < L; ++d) {
    const float t = (float)(d - WR);
    kv[d] = (-(t * t) / S2) * coefc;
  }

  const float* xb = x + (size_t)b * STRIDE_B + SCALE * C_ + c;

  // ---- Pad columns of T: exactly mn (see note above) ----
  for (int i = tid; i < WR * C_; i += 256) {
    T[i] = mn;                    // left pad  (wq in [0, WR))
    T[(WR + W_) * C_ + i] = mn;   // right pad (wq in [WR+W_, WP))
  }

  // ---- Phase 1: dilate along H into LDS for real columns ----
  if (h >= WR && h < H_ - WR) {
    // Interior rows (most blocks): branch-free; taps paired -> v_max3_num_f32.
    const float* xr = xb + (h - WR) * STRIDE_H;
    for (int w = grp; w < W_; w += 8) {
      __builtin_prefetch(xr + WR * STRIDE_H + (w + 8) * STRIDE_W, 0, 1);
      const int col = w * STRIDE_W;
      float a0 = xr[0 * STRIDE_H + col] + kv[0];
      float a1 = xr[1 * STRIDE_H + col] + kv[1];
#pragma unroll
      for (int d = 2; d + 1 < L; d += 2) {
        const float v0 = xr[d * STRIDE_H + col] + kv[d];
        const float v1 = xr[(d + 1) * STRIDE_H + col] + kv[d + 1];
        if (d & 2) a1 = max3f(a1, v0, v1);
        else       a0 = max3f(a0, v0, v1);
      }
      const float vl = xr[(L - 1) * STRIDE_H + col] + kv[L - 1];
      T[(WR + w) * C_ + c] = max3f(a0, a1, vl);
    }
  } else {
    // Boundary rows (2*WR per scale): skip dominated out-of-range taps.
    for (int w = grp; w < W_; w += 8) {
      const int col = w * STRIDE_W;
      float a0 = mn, a1 = mn, a2 = mn, a3 = mn;
#pragma unroll
      for (int d = 0; d < L; ++d) {
        const int row = h + d - WR;            // uniform per block -> scalar branch
        if (row >= 0 && row < H_) {
          const float v = xb[row * STRIDE_H + col] + kv[d];
          if ((d & 3) == 0) a0 = fmaxf(a0, v);
          else if ((d & 3) == 1) a1 = fmaxf(a1, v);
          else if ((d & 3) == 2) a2 = fmaxf(a2, v);
          else a3 = fmaxf(a3, v);
        }
      }
      T[(WR + w) * C_ + c] = fmaxf(fmaxf(a0, a1), fmaxf(a2, a3));
    }
  }
  __syncthreads();

  // ---- Phase 2: dilate along W from LDS, store coalesced ----
  float* ob = out + (size_t)b * STRIDE_B + (size_t)h * STRIDE_H + SCALE * C_ + c;
  for (int m = grp; m < W_; m += 8) {
    const int tbase = m * C_ + c;
    float a0 = T[tbase + 0 * C_] + kv[0];
    float a1 = T[tbase + 1 * C_] + kv[1];
#pragma unroll
    for (int d = 2; d + 1 < L; d += 2) {
      const float v0 = T[tbase + d * C_] + kv[d];
      const float v1 = T[tbase + (d + 1) * C_] + kv[d + 1];
      if (d & 2) a1 = max3f(a1, v0, v1);
      else       a0 = max3f(a0, v0, v1);
    }
    const float vl = T[tbase + (L - 1) * C_] + kv[L - 1];
    ob[m * STRIDE_W] = max3f(a0, a1, vl);
  }
}

extern "C" void kernel_launch(void* const* d_in, const int* in_sizes, int n_in,
                              void* d_out, int out_size, void* d_ws, size_t ws_size,
                              hipStream_t stream) {
  (void)in_sizes; (void)n_in; (void)out_size; (void)ws_size;
  const float* x = (const float*)d_in[0];     // (8,256,256,3,32) f32
  const float* coef = (const float*)d_in[1];  // (1,32) f32
  float* out = (float*)d_out;                 // (8,256,256,3,32) f32
  uint32_t* mins = (uint32_t*)d_ws;           // 3 order-encoded minima

  init_mins_kernel<<<1, 32, 0, stream>>>(mins);
  slice_min_kernel<<<2048, 256, 0, stream>>>((const float4*)x, mins);

  dim3 grid(H_, B_);
  dilate2d_kernel<5, 0><<<grid, 256, 0, stream>>>(x, coef, mins, out);
  dilate2d_kernel<10, 1><<<grid, 256, 0, stream>>>(x, coef, mins, out);
  dilate2d_kernel<20, 2><<<grid, 256, 0, stream>>>(x, coef, mins, out);
}